// ConsolidationDynamics_70068096467285
// MI455X (gfx1250) — compile-verified
//
#include <hip/hip_runtime.h>

// CDNA5 / gfx1250: elementwise tiny-MLP "consolidation dynamics".
// Layer 1 (16x4 @ 4x16 f32, bias folded into K=3) runs on the matrix pipe via
// V_WMMA_F32_16X16X4_F32; layer 2 is an in-lane 8-term FMA dot + ONE
// ds_swizzle SWAPX16. ReLU/clamp via v_med3_f32 (finite bounds so the compiler
// can't re-expand to canonicalize+max); tanh uses v_exp + fast v_rcp.
// All loop bounds are scalar kernargs -> pure SALU control flow, EXEC all-ones
// around WMMA as the ISA requires. Streaming kernel; HBM roofline ~5.8us for
// 2^24 f32 in + out at 23.3 TB/s.

typedef __attribute__((ext_vector_type(2))) float v2f;
typedef __attribute__((ext_vector_type(8))) float v8f;

#define CONSOLIDATION_RATE 0.001f
#define CLAMP_V 10.0f
#define FLT_BIG 3.38953139e38f   // < FLT_MAX; finite med3 bound (not foldable to fmax)
#define SWAPX16_IMM 0x401F       // ds_swizzle group-of-32: xor=0x10, or=0, and=0x1f

__device__ __forceinline__ float swz16(float x) {
    // swap lanes 0-15 <-> 16-31 (wave32), single ds_swizzle_b32, no index VGPR
    return __int_as_float(__builtin_amdgcn_ds_swizzle(__float_as_int(x), SWAPX16_IMM));
}

__device__ __forceinline__ float relu1(float x) {
    // single v_med3_f32: med(x, 0, big) == relu for all finite pre-activations
    return __builtin_amdgcn_fmed3f(x, 0.0f, FLT_BIG);
}

__device__ __forceinline__ float tanh_fast(float x) {
    // tanh(x) = 1 - 2/(exp(2x)+1); exp->inf gives 1, exp->0 gives -1 (no NaN)
    float e = __expf(2.0f * x);
    return 1.0f - 2.0f * __builtin_amdgcn_rcpf(e + 1.0f);
}

struct Ctx {
    v2f   A;        // W1^T rows for this lane (K0/K1 low half, K2/bias high half)
    float bK13;     // B VGPR1: cs (lanes 0-15) / 1.0 (lanes 16-31)
    float fs;       // B VGPR0 high-half constant (K=2 row)
    float b2;
    float w2c[8];   // layer-2 coefficient for D VGPR r (j=r low half, j=r+8 high)
    bool  lo;
};

__device__ __forceinline__ float mlp_group32(float w, const Ctx& c) {
    // One 32-element group = two 16-element WMMA tiles.
    float wsw = swz16(w);                       // high-half w's into low lanes
    v2f B0, B1;
    B0.x = c.lo ? w   : c.fs;  B0.y = c.bK13;   // tile 0: elements [base, base+16)
    B1.x = c.lo ? wsw : c.fs;  B1.y = c.bK13;   // tile 1: elements [base+16, base+32)
    v8f cz = {};
    v8f d0 = __builtin_amdgcn_wmma_f32_16x16x4_f32(false, c.A, false, B0,
                                                   (short)0, cz, false, false);
    v8f d1 = __builtin_amdgcn_wmma_f32_16x16x4_f32(false, c.A, false, B1,
                                                   (short)0, cz, false, false);
    float p0 = 0.0f, p1 = 0.0f;
#pragma unroll
    for (int r = 0; r < 8; ++r) {
        p0 = fmaf(relu1(d0[r]), c.w2c[r], p0);
        p1 = fmaf(relu1(d1[r]), c.w2c[r], p1);
    }
    // lane l<16 needs p0[l^16]; lane l>=16 needs p1[l^16]: one swizzle serves both.
    float own = c.lo ? p0 : p1;
    float oth = swz16(c.lo ? p1 : p0);
    float u   = own + oth + c.b2;
    float nw  = fmaf(tanh_fast(u), CONSOLIDATION_RATE, w);
    return __builtin_amdgcn_fmed3f(nw, -CLAMP_V, CLAMP_V);   // single-instr clamp
}

// Requires: total threads == stride, stride % 32 == 0, n == stride * iters.
// All control flow depends only on scalar kernargs -> no exec masking anywhere.
__global__ __launch_bounds__(256)
void ConsolidationDynamics_wmma_kernel(const float* __restrict__ w_in,
                                       const float* __restrict__ cs_p,
                                       const float* __restrict__ fs_p,
                                       const float* __restrict__ W1,   // 3x16 row-major
                                       const float* __restrict__ b1,   // 16
                                       const float* __restrict__ W2,   // 16x1
                                       const float* __restrict__ b2_p, // 1
                                       float* __restrict__ w_out,
                                       int stride, int iters) {
    const int lane = threadIdx.x & 31;
    Ctx c;
    c.lo = lane < 16;
    const int j = lane & 15;

    const float cs = cs_p[0];
    c.fs = fs_p[0];
    c.b2 = b2_p[0];

    // A-matrix (16x4 f32, M=hidden j, K=input), loop-invariant:
    //   lanes 0-15 : VGPR0=K0=W1[0,j], VGPR1=K1=W1[1,j]
    //   lanes 16-31: VGPR0=K2=W1[2,j], VGPR1=K3=b1[j]  (input K3 row == 1.0)
    c.A.x = c.lo ? W1[0 * 16 + j] : W1[2 * 16 + j];
    c.A.y = c.lo ? W1[1 * 16 + j] : b1[j];
    c.bK13 = c.lo ? cs : 1.0f;
#pragma unroll
    for (int r = 0; r < 8; ++r) c.w2c[r] = c.lo ? W2[r] : W2[r + 8];

    int idx = blockIdx.x * blockDim.x + threadIdx.x;

    // Pairs of groups for ILP (two independent load->WMMA->store chains).
    int k = 0;
    for (; k + 1 < iters; k += 2) {
        float w0 = w_in[idx];
        float w1 = w_in[idx + stride];
        __builtin_prefetch(w_in + idx + 2 * stride, 0, 3);
        w_out[idx]          = mlp_group32(w0, c);
        w_out[idx + stride] = mlp_group32(w1, c);
        idx += 2 * stride;
    }
    if (k < iters) {                       // scalar (kernarg-derived) condition
        float w0 = w_in[idx];
        w_out[idx] = mlp_group32(w0, c);
    }
}

// Scalar VALU fallback for sizes without a clean stride decomposition.
__global__ __launch_bounds__(256)
void ConsolidationDynamics_scalar_kernel(const float* __restrict__ w_in,
                                         const float* __restrict__ cs_p,
                                         const float* __restrict__ fs_p,
                                         const float* __restrict__ W1,
                                         const float* __restrict__ b1,
                                         const float* __restrict__ W2,
                                         const float* __restrict__ b2_p,
                                         float* __restrict__ w_out,
                                         int n) {
    int i = blockIdx.x * blockDim.x + threadIdx.x;
    if (i >= n) return;
    float w  = w_in[i];
    float cs = cs_p[0], fs = fs_p[0];
    float u  = b2_p[0];
#pragma unroll
    for (int h = 0; h < 16; ++h) {
        float pre = fmaf(w, W1[h], fmaf(cs, W1[16 + h], fmaf(fs, W1[32 + h], b1[h])));
        u = fmaf(relu1(pre), W2[h], u);
    }
    float nw = fmaf(tanh_fast(u), CONSOLIDATION_RATE, w);
    w_out[i] = __builtin_amdgcn_fmed3f(nw, -CLAMP_V, CLAMP_V);
}

extern "C" void kernel_launch(void* const* d_in, const int* in_sizes, int n_in,
                              void* d_out, int out_size, void* d_ws, size_t ws_size,
                              hipStream_t stream) {
    const float* w_in = (const float*)d_in[0];
    const float* cs   = (const float*)d_in[1];
    const float* fs   = (const float*)d_in[2];
    const float* W1   = (const float*)d_in[3];
    const float* b1   = (const float*)d_in[4];
    const float* W2   = (const float*)d_in[5];
    const float* b2   = (const float*)d_in[6];
    float* out        = (float*)d_out;
    const int n = in_sizes[0];

    const int block = 256;   // 8 wave32's per workgroup

    // Find iters (grid-stride trip count) with n == (blocks*256) * iters exactly,
    // preferring ~8 groups per wave. For n = 4096*4096: blocks=8192, iters=8.
    int launched = 0;
    const int iter_opts[] = {8, 4, 2, 16, 1, 32, 64};
    for (int t = 0; t < 7 && !launched; ++t) {
        const long long chunk = (long long)block * iter_opts[t];
        if (n > 0 && (n % chunk) == 0) {
            int blocks = (int)(n / chunk);
            int stride = blocks * block;           // multiple of 32
            ConsolidationDynamics_wmma_kernel<<<blocks, block, 0, stream>>>(
                w_in, cs, fs, W1, b1, W2, b2, out, stride, iter_opts[t]);
            launched = 1;
        }
    }
    if (!launched) {
        int blocks = (n + block - 1) / block;
        if (blocks < 1) blocks = 1;
        ConsolidationDynamics_scalar_kernel<<<blocks, block, 0, stream>>>(
            w_in, cs, fs, W1, b1, W2, b2, out, n);
    }
}